// MambaSSM_47476568490196
// MI455X (gfx1250) — compile-verified
//
#include <hip/hip_runtime.h>
#include <hip/hip_bf16.h>
#include <math.h>

// ---------------------------------------------------------------------------
// MambaSSM pipeline for MI455X (gfx1250, wave32).
//   All GEMMs use V_WMMA_F32_16X16X4_F32 (full f32 precision; compute is
//   ~2.7 GFLOP, totally negligible vs the 268 MB output write, so we keep
//   max precision). Final "row replication" is algebraically hoisted:
//   out rows for i>=1 are 256 copies of z[i] where z = u @ W_out^T + b_out,
//   so we compute z (1 MB) once and stream 268 MB of b128 NT stores.
// ---------------------------------------------------------------------------

typedef float v2f __attribute__((ext_vector_type(2)));
typedef float v4f __attribute__((ext_vector_type(4)));
typedef float v8f __attribute__((ext_vector_type(8)));

#define T_SEQ 1024
#define H_DIM 256

// ---------------------------------------------------------------------------
// GEMM: Out[M,N] = act(A[M,K] @ W[N,K]^T + bias[N]) (+ dvec[n]*hmat[m,n])
// One wave per 16x16 tile, K stepped by 4 via v_wmma_f32_16x16x4_f32.
// ACT: 0 = none, 1 = exact GELU, 2 = sigmoid.  bias may be nullptr (=> 0).
// ---------------------------------------------------------------------------
template <int ACT, bool FUSE>
__global__ __launch_bounds__(256) void gemm_nt_wmma(
    const float* __restrict__ A, const float* __restrict__ W,
    const float* __restrict__ bias, float* __restrict__ Out,
    int M, int N, int K,
    const float* __restrict__ dvec, const float* __restrict__ hmat)
{
    const int wid    = blockIdx.x * (blockDim.x >> 5) + (threadIdx.x >> 5);
    const int lane   = threadIdx.x & 31;
    const int tilesN = N >> 4;
    const int tm     = (wid / tilesN) << 4;
    const int tn     = (wid % tilesN) << 4;

    // A-frag: lane<16 holds (m=lane, k0..k0+1); lane>=16 holds (m=lane-16, k0+2..k0+3)
    // B-frag: same pattern with n in place of m; B[k][n] = W[n][k] (contiguous in k).
    const int r     = lane & 15;
    const int khalf = (lane >> 4) << 1;           // 0 or 2
    const float* arow = A + (size_t)(tm + r) * K + khalf;
    const float* brow = W + (size_t)(tn + r) * K + khalf;

    v8f acc = {};
    for (int k0 = 0; k0 < K; k0 += 4) {
        v2f a = *(const v2f*)(arow + k0);
        v2f b = *(const v2f*)(brow + k0);
        acc = __builtin_amdgcn_wmma_f32_16x16x4_f32(
            /*neg_a=*/false, a, /*neg_b=*/false, b,
            /*c_mod=*/(short)0, acc, /*reuse_a=*/false, /*reuse_b=*/false);
    }

    // C/D layout: VGPR rr -> M = rr (lanes 0-15) or rr+8 (lanes 16-31); N = lane&15.
    const int   n    = tn + (lane & 15);
    const int   moff = (lane >> 4) << 3;          // 0 or 8
    const float bn   = bias ? bias[n] : 0.0f;     // uniform branch, post-WMMA
#pragma unroll
    for (int rr = 0; rr < 8; ++rr) {
        const int m = tm + moff + rr;
        float v = acc[rr] + bn;
        if (ACT == 1) v = 0.5f * v * (1.0f + erff(v * 0.70710678118654752f));
        if (ACT == 2) v = 1.0f / (1.0f + __expf(-v));
        if (FUSE)     v += dvec[n] * hmat[(size_t)m * N + n];
        Out[(size_t)m * N + n] = v;
    }
}

// ---------------------------------------------------------------------------
// h[row] = LayerNorm(y[row]) * gamma + beta + h[row]   (one block per row)
// ---------------------------------------------------------------------------
__global__ __launch_bounds__(256) void ln_residual_kernel(
    const float* __restrict__ y, const float* __restrict__ gamma,
    const float* __restrict__ beta, float* __restrict__ h)
{
    __shared__ float red[256];
    const int row = blockIdx.x;
    const int t   = threadIdx.x;
    const float val = y[row * H_DIM + t];

    red[t] = val; __syncthreads();
    for (int s = 128; s > 0; s >>= 1) {
        if (t < s) red[t] += red[t + s];
        __syncthreads();
    }
    const float mean = red[0] * (1.0f / H_DIM);
    __syncthreads();

    const float d = val - mean;
    red[t] = d * d; __syncthreads();
    for (int s = 128; s > 0; s >>= 1) {
        if (t < s) red[t] += red[t + s];
        __syncthreads();
    }
    const float var = red[0] * (1.0f / H_DIM);

    h[row * H_DIM + t] = d * rsqrtf(var + 1e-5f) * gamma[t] + beta[t]
                       + h[row * H_DIM + t];
}

// ---------------------------------------------------------------------------
// v_0 = h_0 ; v_t = dt_t * v_{t-1} + (1-dt_t) * h_t  (parallel over H channels)
// ---------------------------------------------------------------------------
__global__ __launch_bounds__(256) void scan_kernel(
    const float* __restrict__ h, const float* __restrict__ dt,
    float* __restrict__ vs)
{
    const int c = threadIdx.x;
    float v = h[c];
    vs[c] = v;
    for (int t = 1; t < T_SEQ; ++t) {
        const float d  = dt[t * H_DIM + c];
        const float hh = h[t * H_DIM + c];
        v = d * v + (1.0f - d) * hh;
        vs[t * H_DIM + c] = v;
    }
}

// ---------------------------------------------------------------------------
// out row 0 = z row 0; out rows [1+(i-1)*256, 1+i*256) = z row i, i=1..1023.
// Pure bandwidth kernel: 128-bit non-temporal stores (268 MB > 192 MB L2).
// z is 1 MB and L2-resident for the re-reads.
// ---------------------------------------------------------------------------
__global__ __launch_bounds__(256) void bcast_out_kernel(
    const float* __restrict__ z, float* __restrict__ out, int total4)
{
    int g = blockIdx.x * blockDim.x + threadIdx.x;
    const int stride = gridDim.x * blockDim.x;
    for (; g < total4; g += stride) {
        const int row  = g >> 6;                       // 64 float4 per 256-f row
        const int col4 = g & 63;
        const int src  = (row == 0) ? 0 : (1 + ((row - 1) >> 8));
        const v4f v = ((const v4f*)(z + (size_t)src * H_DIM))[col4];
        __builtin_nontemporal_store(v, ((v4f*)out) + g);
    }
}

// ---------------------------------------------------------------------------
extern "C" void kernel_launch(void* const* d_in, const int* in_sizes, int n_in,
                              void* d_out, int out_size, void* d_ws, size_t ws_size,
                              hipStream_t stream) {
    (void)in_sizes; (void)n_in; (void)ws_size;
    const float* x     = (const float*)d_in[0];
    const float* W_in  = (const float*)d_in[1];
    const float* b_in  = (const float*)d_in[2];
    const float* W1    = (const float*)d_in[3];   // (L, 2H, H)
    const float* b1    = (const float*)d_in[4];   // (L, 2H)
    const float* W2    = (const float*)d_in[5];   // (L, H, 2H)
    const float* b2    = (const float*)d_in[6];   // (L, H)
    const float* gamma = (const float*)d_in[7];   // (L, H)
    const float* beta  = (const float*)d_in[8];   // (L, H)
    const float* W_dt  = (const float*)d_in[9];
    const float* b_dt  = (const float*)d_in[10];
    const float* Cmat  = (const float*)d_in[11];  // (H, S)
    const float* Dvec  = (const float*)d_in[12];  // (H,)
    const float* W_out = (const float*)d_in[13];
    const float* b_out = (const float*)d_in[14];
    float* out = (float*)d_out;

    const int T = T_SEQ, H = H_DIM, H2 = 2 * H_DIM, L = 4;

    float* ws = (float*)d_ws;
    float* h  = ws;                 // T*H
    float* y1 = h  + T * H;         // T*2H
    float* y2 = y1 + T * H2;        // T*H
    float* dt = y2 + T * H;         // T*H
    float* vs = dt + T * H;         // T*H
    float* u  = vs + T * H;         // T*H
    float* z  = u  + T * H;         // T*H  (total ~8 MB)

    auto nblk = [](int M, int N) { return (M >> 4) * (N >> 4) / 8; };  // 8 waves/block

    // h = x @ W_in^T + b_in
    gemm_nt_wmma<0, false><<<nblk(T, H), 256, 0, stream>>>(
        x, W_in, b_in, h, T, H, H, nullptr, nullptr);

    // residual gated-MLP stack
    for (int l = 0; l < L; ++l) {
        gemm_nt_wmma<1, false><<<nblk(T, H2), 256, 0, stream>>>(
            h, W1 + (size_t)l * H2 * H, b1 + (size_t)l * H2, y1,
            T, H2, H, nullptr, nullptr);
        gemm_nt_wmma<0, false><<<nblk(T, H), 256, 0, stream>>>(
            y1, W2 + (size_t)l * H * H2, b2 + (size_t)l * H, y2,
            T, H, H2, nullptr, nullptr);
        ln_residual_kernel<<<T, 256, 0, stream>>>(
            y2, gamma + (size_t)l * H, beta + (size_t)l * H, h);
    }

    // dt = sigmoid(h @ W_dt^T + b_dt)
    gemm_nt_wmma<2, false><<<nblk(T, H), 256, 0, stream>>>(
        h, W_dt, b_dt, dt, T, H, H, nullptr, nullptr);

    // channel-wise recurrence
    scan_kernel<<<1, 256, 0, stream>>>(h, dt, vs);

    // u = vs @ C^T + D * h   (fused epilogue, no bias term)
    gemm_nt_wmma<0, true><<<nblk(T, H), 256, 0, stream>>>(
        vs, Cmat, nullptr, u, T, H, H, Dvec, h);

    // z = u @ W_out^T + b_out
    gemm_nt_wmma<0, false><<<nblk(T, H), 256, 0, stream>>>(
        u, W_out, b_out, z, T, H, H, nullptr, nullptr);

    // broadcast-replicate rows into the 268 MB output
    const int total4 = out_size / 4;
    bcast_out_kernel<<<8192, 256, 0, stream>>>(z, out, total4);
}